// CrossSubgConv_3496103379079
// MI455X (gfx1250) — compile-verified
//
#include <hip/hip_runtime.h>

typedef float v2f __attribute__((ext_vector_type(2)));
typedef float v4f __attribute__((ext_vector_type(4)));
typedef float v8f __attribute__((ext_vector_type(8)));

// D = A(16x4 f32) * B(4x16 f32) + C(16x16 f32)
#define WMMA_F32(a, b, c) \
  __builtin_amdgcn_wmma_f32_16x16x4_f32(false, (a), false, (b), (short)0, (c), false, false)

constexpr int Bn = 32, Nn = 128, Dn = 64;
// Padded LDS row stride for 64-wide tiles. Even (8B-aligned f32 pairs -> single
// ds_load_b64 per WMMA A-fragment) and 66 mod 64 == 2, so column reads across
// 16 consecutive rows land on distinct bank pairs (no serialization).
constexpr int XS = 66;

__global__ __launch_bounds__(256, 1)
void cross_subg_conv_kernel(const float* __restrict__ X,
                            const float* __restrict__ A,
                            const float* __restrict__ W1,
                            const float* __restrict__ b1,
                            const float* __restrict__ W2,
                            const float* __restrict__ b2,
                            float* __restrict__ out)
{
    // sX holds the X tile [k][d]; reused as H2 after GEMM2.
    __shared__ float sX[Nn * XS];
    __shared__ float sH1[Nn * XS];
    __shared__ float sW[2 * Dn * XS];   // W1 then W2, row-major [d][e]
    __shared__ float sBias[2 * Dn];

    const int i    = blockIdx.x;
    const int b    = blockIdx.y;
    const int tid  = threadIdx.x;
    const int lane = tid & 31;
    const int wave = tid >> 5;
    const int half = lane >> 4;   // 0: lanes 0-15 (K=k0,k0+1), 1: lanes 16-31 (K=k0+2,k0+3)
    const int lr   = lane & 15;
    const int mBase = wave * 16;

    // ---- stage X tile + weights + biases into LDS ----
    // X is streamed exactly once across the whole dispatch: non-temporal loads
    // keep the 134 MB stream from evicting the L2-resident A / weight data.
    const v4f* Xg = reinterpret_cast<const v4f*>(X + (size_t)(b * Nn + i) * Nn * Dn);
    for (int t = tid; t < Nn * Dn / 4; t += 256) {
        v4f v = __builtin_nontemporal_load(Xg + t);
        int row = t >> 4, col = (t & 15) << 2;
        *reinterpret_cast<v4f*>(&sX[row * XS + col]) = v;   // (row*66+col)*4 is 16B-aligned? col%4==0, row*66 even -> 8B; store as v4f ok unaligned-8? keep scalar-safe below
    }
    const v4f* W1g = reinterpret_cast<const v4f*>(W1);
    const v4f* W2g = reinterpret_cast<const v4f*>(W2);
    for (int t = tid; t < Dn * Dn / 4; t += 256) {
        int row = t >> 4, col = (t & 15) << 2;
        v4f v1 = W1g[t];
        v4f v2 = W2g[t];
        *reinterpret_cast<v2f*>(&sW[row * XS + col])          = v1.xy;
        *reinterpret_cast<v2f*>(&sW[row * XS + col + 2])      = v1.zw;
        *reinterpret_cast<v2f*>(&sW[Dn * XS + row * XS + col])     = v2.xy;
        *reinterpret_cast<v2f*>(&sW[Dn * XS + row * XS + col + 2]) = v2.zw;
    }
    if (tid < Dn) { sBias[tid] = b1[tid]; sBias[Dn + tid] = b2[tid]; }
    __syncthreads();

    // ---- GEMM1: H1 = relu(X @ W1 + b1)   (128x64 = 128x64 * 64x64) ----
    {
        const int rowBase = (mBase + lr) * XS;
        v8f acc[4] = {};
        #pragma unroll
        for (int kk = 0; kk < 16; ++kk) {
            const int k0 = kk * 4 + half * 2;
            v2f a = *reinterpret_cast<const v2f*>(&sX[rowBase + k0]);  // one ds_load_b64
            #pragma unroll
            for (int nt = 0; nt < 4; ++nt) {
                v2f bf;
                bf.x = sW[k0 * XS + nt * 16 + lr];
                bf.y = sW[(k0 + 1) * XS + nt * 16 + lr];
                acc[nt] = WMMA_F32(a, bf, acc[nt]);
            }
        }
        #pragma unroll
        for (int nt = 0; nt < 4; ++nt) {
            const float bias = sBias[nt * 16 + lr];
            #pragma unroll
            for (int v = 0; v < 8; ++v) {
                float r = fmaxf(acc[nt][v] + bias, 0.0f);
                sH1[(mBase + v + half * 8) * XS + nt * 16 + lr] = r;
            }
        }
    }
    __syncthreads();  // GEMM1 reads of sX done; sH1 fully written

    // ---- GEMM2: H2 = relu(H1 @ W2 + b2) -> written into sX ----
    {
        const int rowBase = (mBase + lr) * XS;
        v8f acc[4] = {};
        #pragma unroll
        for (int kk = 0; kk < 16; ++kk) {
            const int k0 = kk * 4 + half * 2;
            v2f a = *reinterpret_cast<const v2f*>(&sH1[rowBase + k0]);
            #pragma unroll
            for (int nt = 0; nt < 4; ++nt) {
                v2f bf;
                bf.x = sW[Dn * XS + k0 * XS + nt * 16 + lr];
                bf.y = sW[Dn * XS + (k0 + 1) * XS + nt * 16 + lr];
                acc[nt] = WMMA_F32(a, bf, acc[nt]);
            }
        }
        #pragma unroll
        for (int nt = 0; nt < 4; ++nt) {
            const float bias = sBias[Dn + nt * 16 + lr];
            #pragma unroll
            for (int v = 0; v < 8; ++v) {
                float r = fmaxf(acc[nt][v] + bias, 0.0f);
                sX[(mBase + v + half * 8) * XS + nt * 16 + lr] = r;
            }
        }
    }
    __syncthreads();  // sX now holds H2

    // ---- GEMM3: out[j,d] = sum_k A[b,k,j] * H2[k,d]   (A^T @ H2, 128x128x64) ----
    {
        const float* Ab = A + (size_t)b * Nn * Nn;   // [k][j]; 2 MB total, reused 128x -> L2-resident
        v8f acc[4] = {};
        #pragma unroll 8
        for (int kk = 0; kk < 32; ++kk) {
            const int k0 = kk * 4 + half * 2;
            v2f a;
            a.x = Ab[k0 * Nn + mBase + lr];          // A^T[j,k0]   = A[k0, j]
            a.y = Ab[(k0 + 1) * Nn + mBase + lr];    // A^T[j,k0+1] = A[k0+1, j]
            #pragma unroll
            for (int nt = 0; nt < 4; ++nt) {
                v2f bf;
                bf.x = sX[k0 * XS + nt * 16 + lr];
                bf.y = sX[(k0 + 1) * XS + nt * 16 + lr];
                acc[nt] = WMMA_F32(a, bf, acc[nt]);
            }
        }
        // Output is written exactly once: non-temporal stores keep the 134 MB
        // stream out of the reused L2 working set.
        float* og = out + (size_t)(b * Nn + i) * Nn * Dn;
        #pragma unroll
        for (int nt = 0; nt < 4; ++nt) {
            #pragma unroll
            for (int v = 0; v < 8; ++v) {
                __builtin_nontemporal_store(
                    acc[nt][v],
                    &og[(size_t)(mBase + v + half * 8) * Dn + nt * 16 + lr]);
            }
        }
    }
}

extern "C" void kernel_launch(void* const* d_in, const int* in_sizes, int n_in,
                              void* d_out, int out_size, void* d_ws, size_t ws_size,
                              hipStream_t stream) {
    const float* X  = (const float*)d_in[0];
    const float* A  = (const float*)d_in[1];
    const float* W1 = (const float*)d_in[2];
    const float* b1 = (const float*)d_in[3];
    const float* W2 = (const float*)d_in[4];
    const float* b2 = (const float*)d_in[5];
    float* out = (float*)d_out;

    dim3 grid(Nn, Bn);   // one workgroup per (b, i)
    dim3 block(256);     // 8 waves (wave32)
    cross_subg_conv_kernel<<<grid, block, 0, stream>>>(X, A, W1, b1, W2, b2, out);
}